// LogWeightedDICELossMultiClass3D_20658792693789
// MI455X (gfx1250) — compile-verified
//
#include <hip/hip_runtime.h>

typedef __attribute__((ext_vector_type(2))) float v2f;
typedef __attribute__((ext_vector_type(4))) float v4f;
typedef __attribute__((ext_vector_type(8))) float v8f;

#define B_ 4
#define C_ 3
#define Z_ 64
#define H_ 192
#define W_ 192
#define W4_ (W_ / 4)       // 48 float4 groups per row
#define TH 32              // h-rows per block tile
#define NHT (H_ / TH)      // 6 tiles per slice
#define NSUM 5
#define FIXSCALE 1048576.0 // 2^20 fixed-point for deterministic integer atomics

// ---------------------------------------------------------------------------
// ws layout: unsigned long long sums[B_*C_][NSUM]
//   0: sum(mask)  1: sum(ts==mask)  2: sum(edge)  3: sum(p*m)  4: sum(p)
// all scaled by 2^20 (counts exact; p-sums quantized only at block granularity)
// ---------------------------------------------------------------------------

__global__ void lwd_zero_ws(unsigned long long* ws) {
    int i = threadIdx.x;
    if (i < B_ * C_ * NSUM) ws[i] = 0ULL;
}

__global__ __launch_bounds__(256) void lwd_partial_kernel(
    const float* __restrict__ outp, const float* __restrict__ mskp,
    const float* __restrict__ thr_p, unsigned long long* __restrict__ ws)
{
    // z-smoothed thresholded segmentation, rows h0-1 .. h0+TH (reflect), full W
    __shared__ float s0[(TH + 2) * W_];
    // per-thread partial sums staged as a [256][16] matrix for WMMA reduction
    __shared__ float red[256 * 16];

    const float thr = thr_p[0];
    const int tid = threadIdx.x;
    int blk = blockIdx.x;
    const int ht = blk % NHT; blk /= NHT;
    const int z  = blk % Z_;  blk /= Z_;
    const int bc = blk;                 // 0 .. B_*C_-1
    const int h0 = ht * TH;

    const size_t vbase = (size_t)bc * ((size_t)Z_ * H_ * W_);
    const int zm = (z == 0) ? 0 : z - 1;              // reflect ('symmetric')
    const int zp = (z == Z_ - 1) ? Z_ - 1 : z + 1;
    const float* pm = outp + vbase + (size_t)zm * (H_ * W_);
    const float* pc = outp + vbase + (size_t)z  * (H_ * W_);
    const float* pp = outp + vbase + (size_t)zp * (H_ * W_);
    const float* mc = mskp + vbase + (size_t)z  * (H_ * W_);

    // ---- Phase 1: s0 = ts(zm)+2*ts(z)+ts(zp), reflect rows, float4 groups --
    for (int g = tid; g < (TH + 2) * W4_; g += 256) {
        int r  = g / W4_;
        int w4 = g - r * W4_;
        int hr = h0 - 1 + r;
        hr = (hr < 0) ? 0 : ((hr >= H_) ? H_ - 1 : hr);
        size_t o = (size_t)hr * W_ + 4 * w4;
        v4f ta = *(const v4f*)(pm + o);      // global_load_b128
        v4f tb = *(const v4f*)(pc + o);
        v4f tc = *(const v4f*)(pp + o);
        v4f s;
        #pragma unroll
        for (int k = 0; k < 4; ++k) {
            s[k] = ((ta[k] > thr) ? 1.0f : 0.0f)
                 + 2.0f * ((tb[k] > thr) ? 1.0f : 0.0f)
                 + ((tc[k] > thr) ? 1.0f : 0.0f);
        }
        *(v4f*)&s0[r * W_ + 4 * w4] = s;     // ds_store_b128 (16B aligned)
    }
    __syncthreads();

    // ---- Phase 2: fused elementwise + sobel-edge + partial sums (float4) --
    float a_m = 0.f, a_eq = 0.f, a_ed = 0.f, a_pm = 0.f, a_p = 0.f;
    #pragma unroll
    for (int j = 0; j < (TH * W4_) / 256; ++j) {     // 6 iterations
        int g  = tid + j * 256;
        int hh = g / W4_;                    // 0..TH-1
        int w4 = g - hh * W4_;
        int r  = hh + 1;                     // center row within s0 tile
        // clamped side columns implement 'symmetric' reflect at w edges
        int wl = (w4 == 0)        ? 0       : 4 * w4 - 1;
        int wr = (w4 == W4_ - 1)  ? W_ - 1  : 4 * w4 + 4;
        const float* r0 = &s0[(r - 1) * W_];
        const float* r1 = &s0[r * W_];
        const float* r2 = &s0[(r + 1) * W_];
        v4f c0 = *(const v4f*)(r0 + 4 * w4); // ds_load_b128 x3
        v4f c1 = *(const v4f*)(r1 + 4 * w4);
        v4f c2 = *(const v4f*)(r2 + 4 * w4);
        v4f s1c = c0 + 2.0f * c1 + c2;       // h-smoothed, 4 center columns
        float s1l = r0[wl] + 2.0f * r1[wl] + r2[wl];
        float s1r = r0[wr] + 2.0f * r1[wr] + r2[wr];
        // grad[w] = s1[w+1] - s1[w-1] (reflect); edge = grad > 0 (exact ints)
        a_ed += ((s1c[1] - s1l)    > 0.0f) ? 1.0f : 0.0f;
        a_ed += ((s1c[2] - s1c[0]) > 0.0f) ? 1.0f : 0.0f;
        a_ed += ((s1c[3] - s1c[1]) > 0.0f) ? 1.0f : 0.0f;
        a_ed += ((s1r    - s1c[2]) > 0.0f) ? 1.0f : 0.0f;

        size_t o = (size_t)(h0 + hh) * W_ + 4 * w4;
        v4f p = *(const v4f*)(pc + o);       // global_load_b128 x2
        v4f m = *(const v4f*)(mc + o);
        #pragma unroll
        for (int k = 0; k < 4; ++k) {
            float ts = (p[k] > thr) ? 1.0f : 0.0f;
            a_eq += (ts == m[k]) ? 1.0f : 0.0f;   // masks are exact 0.0/1.0
            a_pm += p[k] * m[k];
            a_p  += p[k];
            a_m  += m[k];
        }
    }

    // stage partials: row t = [a_m a_eq a_ed a_pm a_p 0 ... 0] (16 cols)
    float* rowp = &red[tid * 16];
    rowp[0] = a_m; rowp[1] = a_eq; rowp[2] = a_ed; rowp[3] = a_pm; rowp[4] = a_p;
    #pragma unroll
    for (int c = NSUM; c < 16; ++c) rowp[c] = 0.0f;
    __syncthreads();

    // ---- Phase 3: WMMA block reduction (wave 0, all 32 lanes active) ------
    // D(16x16) = ones(16x4) x B(4x16) + D : each issue folds 4 threads' rows
    // into per-column sums; 64 chained WMMAs reduce all 256 threads exactly.
    if (tid < 32) {
        const int col  = tid & 15;
        const int half = tid >> 4;
        v2f a; a[0] = 1.0f; a[1] = 1.0f;          // A = ones (16x4 f32)
        v8f d = {};                                // C/D accumulator
        for (int it = 0; it < 64; ++it) {
            // B(4x16): K0=b.x(l<16) K1=b.y(l<16) K2=b.x(l>=16) K3=b.y(l>=16)
            int t0 = it * 4 + half * 2;
            v2f b;
            b[0] = red[t0 * 16 + col];
            b[1] = red[(t0 + 1) * 16 + col];
            d = __builtin_amdgcn_wmma_f32_16x16x4_f32(
                    /*neg_a=*/false, a, /*neg_b=*/false, b,
                    /*c_mod=*/(short)0, d, /*reuse_a=*/false, /*reuse_b=*/false);
        }
        // D row 0, col n lives in VGPR0 of lanes n (n<16); cols 0..4 = totals
        if (tid < NSUM) {
            double v = (double)d[0] * FIXSCALE;    // exact: d[0] is fp32
            atomicAdd(&ws[bc * NSUM + tid], (unsigned long long)llrint(v));
        }
    }
}

__global__ void lwd_finalize_kernel(const unsigned long long* __restrict__ ws,
                                    float* __restrict__ res)
{
    if (threadIdx.x != 0 || blockIdx.x != 0) return;
    const double vox = (double)Z_ * H_ * W_;
    double total = 0.0;
    for (int b = 0; b < B_; ++b) {
        double freq[C_], s_eq[C_], s_ed[C_], s_pm[C_], s_p[C_], s_m[C_];
        for (int c = 0; c < C_; ++c) {
            const unsigned long long* s = &ws[(b * C_ + c) * NSUM];
            s_m[c]  = (double)s[0] / FIXSCALE;
            s_eq[c] = (double)s[1] / FIXSCALE;
            s_ed[c] = (double)s[2] / FIXSCALE;
            s_pm[c] = (double)s[3] / FIXSCALE;
            s_p[c]  = (double)s[4] / FIXSCALE;
            freq[c] = s_m[c] / vox;
        }
        double f0 = freq[0], f1 = freq[1], f2 = freq[2];
        double med = fmax(fmin(f0, f1), fmin(fmax(f0, f1), f2)); // median of 3
        double mn  = fmin(f0, fmin(f1, f2));
        double w0  = 2.0 * med / (mn + 1e-5);
        double nom = 0.0, den = 0.0;
        for (int c = 0; c < C_; ++c) {
            double cw  = s_eq[c] * med / (freq[c] + 1e-5) + w0 * s_ed[c];
            double ps1 = s_pm[c];
            double ps2 = s_p[c] + s_m[c];
            nom += cw * ps1;
            den += cw * ps2 + 1e-7;
        }
        total += 1.0 - 2.0 * nom / den;
    }
    res[0] = (float)(total / (double)B_);
}

extern "C" void kernel_launch(void* const* d_in, const int* in_sizes, int n_in,
                              void* d_out, int out_size, void* d_ws, size_t ws_size,
                              hipStream_t stream) {
    (void)in_sizes; (void)n_in; (void)out_size; (void)ws_size;
    const float* outp = (const float*)d_in[0];
    const float* mskp = (const float*)d_in[1];
    const float* thrp = (const float*)d_in[2];
    unsigned long long* ws = (unsigned long long*)d_ws;   // 60 u64 = 480 B
    float* res = (float*)d_out;

    hipLaunchKernelGGL(lwd_zero_ws, dim3(1), dim3(64), 0, stream, ws);
    const int grid = B_ * C_ * Z_ * NHT;  // 4608 blocks
    hipLaunchKernelGGL(lwd_partial_kernel, dim3(grid), dim3(256), 0, stream,
                       outp, mskp, thrp, ws);
    hipLaunchKernelGGL(lwd_finalize_kernel, dim3(1), dim3(32), 0, stream, ws, res);
}